// Reason_58591943852620
// MI455X (gfx1250) — compile-verified
//
#include <hip/hip_runtime.h>
#include <math.h>

#define B_ROWS 16384
#define R_REL  512
#define V_DICT 1000
#define F_DIM  300
#define L_DIM  512
#define LEAK   0.2f
#define TEMP   0.07f
#define TOPK   2
#define DISK   15
#define KB     32      // K-chunk staged in LDS per double-buffer step

typedef __attribute__((ext_vector_type(2))) float v2f;
typedef __attribute__((ext_vector_type(8))) float v8f;

__device__ __forceinline__ float wave_sum(float v) {
#pragma unroll
  for (int off = 16; off > 0; off >>= 1) v += __shfl_xor(v, off, 32);
  return v;
}

// CDNA5 async global->LDS copies (ASYNCcnt-tracked).
__device__ __forceinline__ void async_g2l_b32(uint32_t lds_off, const float* src) {
  asm volatile("global_load_async_to_lds_b32 %0, %1, off"
               :: "v"(lds_off), "v"(src) : "memory");
}
__device__ __forceinline__ void wait_async0() {
  asm volatile("s_wait_asynccnt 0x0" ::: "memory");
}

// ---------------------------------------------------------------------------
// Zero the two loss scalars and row 0 of z_rel.
__global__ void init_out_kernel(float* __restrict__ out) {
  int i = blockIdx.x * 256 + threadIdx.x;
  if (i < 2 + L_DIM) out[i] = 0.0f;
}

// ---------------------------------------------------------------------------
// Per-row top-2 (desc) and bottom-15 (asc) selection, mapped through rels[:,1].
// One wave32 per row; argsort-stable tie-breaking (smaller index wins).
__global__ void topdis_kernel(const float* __restrict__ logits,
                              const int* __restrict__ rels,
                              int* __restrict__ kk1, int* __restrict__ kk2,
                              int* __restrict__ disK) {
  __shared__ float sbuf[8][R_REL];
  const int w = threadIdx.x >> 5, lane = threadIdx.x & 31;
  const int row = blockIdx.x * 8 + w;
  float* s = sbuf[w];
  const float* lr = logits + (size_t)row * R_REL;
  for (int i = lane; i < R_REL; i += 32) s[i] = lr[i];
  __syncthreads();

  float tv[TOPK]; int ti[TOPK];
  for (int t = 0; t < TOPK; ++t) {
    float bv = -INFINITY; int bi = R_REL;
    for (int i = lane; i < R_REL; i += 32) {
      float v = s[i];
      if (v > bv) { bv = v; bi = i; }          // first hit = smallest idx in-lane
    }
#pragma unroll
    for (int off = 16; off > 0; off >>= 1) {
      float ov = __shfl_xor(bv, off, 32);
      int   oi = __shfl_xor(bi, off, 32);
      if (ov > bv || (ov == bv && oi < bi)) { bv = ov; bi = oi; }
    }
    tv[t] = bv; ti[t] = bi;
    if (lane == 0) {
      int key = rels[2 * bi + 1];
      if (t == 0) kk1[row] = key; else kk2[row] = key;
      s[bi] = -INFINITY;
    }
    __syncthreads();
  }
  // restore top values so they cannot be selected as minima
  if (lane == 0) { s[ti[0]] = tv[0]; s[ti[1]] = tv[1]; }
  __syncthreads();

  for (int t = 0; t < DISK; ++t) {
    float bv = INFINITY; int bi = R_REL;
    for (int i = lane; i < R_REL; i += 32) {
      float v = s[i];
      if (v < bv) { bv = v; bi = i; }
    }
#pragma unroll
    for (int off = 16; off > 0; off >>= 1) {
      float ov = __shfl_xor(bv, off, 32);
      int   oi = __shfl_xor(bi, off, 32);
      if (ov < bv || (ov == bv && oi < bi)) { bv = ov; bi = oi; }
    }
    if (lane == 0) {
      disK[row * DISK + t] = rels[2 * bi + 1];
      s[bi] = INFINITY;
    }
    __syncthreads();
  }
}

// ---------------------------------------------------------------------------
// T[v][j] = sum_f rel_dict[v][f] * W[(f+rowOff)][j]   (V x 512, K = 300)
__global__ void tab_lin_kernel(const float* __restrict__ dict,
                               const float* __restrict__ W, int rowOff,
                               float* __restrict__ out) {
  int e = blockIdx.x * 256 + threadIdx.x;        // V*512
  int v = e >> 9, j = e & 511;
  const float* d = dict + (size_t)v * F_DIM;
  const float* w = W + (size_t)rowOff * L_DIM + j;
  float s = 0.f;
  for (int f = 0; f < F_DIM; ++f) s += d[f] * w[(size_t)f * L_DIM];
  out[e] = s;
}

// hi[v][j] = relu(dict[v] @ iW1 + ib1)
__global__ void imlp1_kernel(const float* __restrict__ dict,
                             const float* __restrict__ iW1,
                             const float* __restrict__ ib1,
                             float* __restrict__ hi) {
  int e = blockIdx.x * 256 + threadIdx.x;
  int v = e >> 9, j = e & 511;
  const float* d = dict + (size_t)v * F_DIM;
  float s = ib1[j];
  for (int f = 0; f < F_DIM; ++f) s += d[f] * iW1[(size_t)f * L_DIM + j];
  hi[e] = fmaxf(s, 0.f);
}

// Tneg[v][j] = relu(hi[v] @ iW2 + ib2)
__global__ void imlp2_kernel(const float* __restrict__ hi,
                             const float* __restrict__ iW2,
                             const float* __restrict__ ib2,
                             float* __restrict__ Tneg) {
  int e = blockIdx.x * 256 + threadIdx.x;
  int v = e >> 9, j = e & 511;
  const float* h = hi + (size_t)v * L_DIM;
  float s = ib2[j];
  for (int f = 0; f < L_DIM; ++f) s += h[f] * iW2[(size_t)f * L_DIM + j];
  Tneg[e] = fmaxf(s, 0.f);
}

// h1[b][j] = leaky(T1a[kk1[b]][j] + T1b[kk2[b]][j] + b1[j])
__global__ void h1_kernel(const int* __restrict__ kk1, const int* __restrict__ kk2,
                          const float* __restrict__ T1a, const float* __restrict__ T1b,
                          const float* __restrict__ b1, float* __restrict__ h1) {
  size_t e = (size_t)blockIdx.x * 256 + threadIdx.x;
  int row = (int)(e >> 9), j = (int)(e & 511);
  float v = T1a[(size_t)kk1[row] * L_DIM + j] + T1b[(size_t)kk2[row] * L_DIM + j] + b1[j];
  h1[e] = v > 0.f ? v : LEAK * v;
}

// z = mu + eps1*exp(ls); zbar = mu + eps2*exp(ls); accumulate KL into klOut
__global__ void z_kernel(const float* __restrict__ mu, const float* __restrict__ ls,
                         const float* __restrict__ eps1, const float* __restrict__ eps2,
                         float* __restrict__ z, float* __restrict__ zbar,
                         float* __restrict__ klOut) {
  size_t e = (size_t)blockIdx.x * 256 + threadIdx.x;
  float m = mu[e], l = ls[e];
  float sd = expf(l);
  z[e]    = fmaf(eps1[e], sd, m);
  zbar[e] = fmaf(eps2[e], sd, m);
  float kl = -0.5f * (1.f + 2.f * l - m * m - expf(2.f * l));
  kl = wave_sum(kl);
  if ((threadIdx.x & 31) == 0) atomicAdd(klOut, kl);
}

// ---------------------------------------------------------------------------
// f32 WMMA GEMM with async-to-LDS staging of the shared W tile.
// Block = 8 waves -> 128 rows x 64 cols; all waves share n0, so the
// W[k..k+KB)[n0..n0+64) tile is staged once per block into double-buffered LDS
// (ASYNCcnt DMA overlapping the WMMAs of the previous chunk).
// LDS layout is K-pair interleaved: wt[pair][col][2] with element (r,c) at
// [r>>1][c][r&1], so the WMMA B fragment (B[ka][c], B[ka+1][c]) is one aligned
// contiguous ds_load_b64 straight into the operand pair (no re-pairing movs).
// B fragments for step i+1 are issued before the WMMAs of step i (partial
// dscnt waits instead of full drains).
// EPI: 0 = +bias                        (mu / log_std)
//      1 = leaky(+bias+gtab[gidx[row]]) (h3: gathered T3b[k2] term)
//      2 = tanh(+bias), squared diff vs gtab[gidx[row]] -> atomic lossOut
//      3 = relu(+bias)                  (z-MLP layers)
template <int EPI, bool GUARD>
__global__ __launch_bounds__(256)
void gemm512_kernel(const float* __restrict__ A, const float* __restrict__ W,
                    const float* __restrict__ bias, float* __restrict__ out,
                    int N, int ldO, int K,
                    const int* __restrict__ gidx,
                    const float* __restrict__ gtab, int gld,
                    float* __restrict__ lossOut) {
  __shared__ __align__(16) float wt[2][KB / 2][64][2];
  const int tid  = threadIdx.x;
  const int lane = tid & 31;
  const int w    = tid >> 5;
  const int m0   = blockIdx.x * 128 + w * 16;
  const int n0   = blockIdx.y * 64;
  const int mr   = lane & 15;
  const int kh   = lane >> 4;
  const float* Arow = A + (size_t)(m0 + mr) * K + 2 * kh;

  // Cooperative async stage: 2048 elems over 256 threads, 8 b32 DMAs each.
  // Global side: 64-wide coalesced rows. LDS side: K-pair interleaved.
  auto stage = [&](int kb, int bi) {
#pragma unroll
    for (int q = 0; q < 8; ++q) {
      int e = tid + q * 256;            // element index in tile
      int r = e >> 6;                   // row 0..KB-1
      int c = e & 63;                   // col 0..63
      int col = n0 + c;
      if (GUARD && col > N - 1) col = N - 1;   // in-bounds; cols >= N discarded
      const float* src = W + (size_t)(kb + r) * N + col;
      async_g2l_b32((uint32_t)(uintptr_t)&wt[bi][r >> 1][c][r & 1], src);
    }
  };

  stage(0, 0);
  wait_async0();
  __syncthreads();

  v8f acc[4] = {};
  const int nchunks = K / KB;
#pragma unroll 2
  for (int ch = 0; ch < nchunks; ++ch) {
    const int cur = ch & 1;
    if (ch + 1 < nchunks) stage((ch + 1) * KB, cur ^ 1);   // overlaps compute below

    v2f areg[KB / 4];
#pragma unroll
    for (int i = 0; i < KB / 4; ++i)
      areg[i] = *reinterpret_cast<const v2f*>(Arow + ch * KB + i * 4);

    // B fragment for k-step i: pair p = i*2 + kh -> (B[4i+2kh][c], B[4i+2kh+1][c])
    v2f bcur[4];
#pragma unroll
    for (int t = 0; t < 4; ++t)
      bcur[t] = *reinterpret_cast<const v2f*>(&wt[cur][kh][t * 16 + mr][0]);

#pragma unroll
    for (int i = 0; i < KB / 4; ++i) {
      v2f bnxt[4];
      if (i + 1 < KB / 4) {
#pragma unroll
        for (int t = 0; t < 4; ++t)
          bnxt[t] = *reinterpret_cast<const v2f*>(
              &wt[cur][(i + 1) * 2 + kh][t * 16 + mr][0]);
      }
#pragma unroll
      for (int t = 0; t < 4; ++t)
        acc[t] = __builtin_amdgcn_wmma_f32_16x16x4_f32(
            false, areg[i], false, bcur[t], (short)0, acc[t], false, false);
      if (i + 1 < KB / 4) {
#pragma unroll
        for (int t = 0; t < 4; ++t) bcur[t] = bnxt[t];
      }
    }
    wait_async0();        // next tile landed (overlapped with the WMMAs above)
    __syncthreads();      // all waves done reading cur before it is restaged
  }

  // Epilogue. C/D layout: VGPR g -> row (lane>>4)*8 + g, col lane&15.
  const int c0    = lane & 15;
  const int rbase = m0 + (lane >> 4) * 8;
  int gi[8];
  if (EPI == 1 || EPI == 2) {
#pragma unroll
    for (int g = 0; g < 8; ++g) gi[g] = gidx[rbase + g];
  }
  float lsum = 0.f;
#pragma unroll
  for (int t = 0; t < 4; ++t) {
    int n = n0 + t * 16 + c0;
    if (GUARD && n >= N) continue;
    float bi = bias[n];
#pragma unroll
    for (int g = 0; g < 8; ++g) {
      int row = rbase + g;
      float v = acc[t][g] + bi;
      if (EPI == 0) {
        out[(size_t)row * ldO + n] = v;
      } else if (EPI == 1) {
        v += gtab[(size_t)gi[g] * gld + n];
        out[(size_t)row * ldO + n] = v > 0.f ? v : LEAK * v;
      } else if (EPI == 2) {
        float tv = tanhf(v);
        float d = tv - gtab[(size_t)gi[g] * gld + n];
        lsum += d * d;
      } else {
        out[(size_t)row * ldO + n] = fmaxf(v, 0.f);
      }
    }
  }
  if (EPI == 2) {
    lsum = wave_sum(lsum);
    if (lane == 0) atomicAdd(lossOut, lsum);
  }
}

// ---------------------------------------------------------------------------
// Contrastive loss: one wave per row. s0 = <z_p, Tneg[kk1]>/T, s1..15 from
// disK gathers; loss += logsumexp(s) - s0.
__global__ void csl_kernel(const float* __restrict__ zp, const float* __restrict__ Tneg,
                           const int* __restrict__ kk1, const int* __restrict__ disK,
                           float* __restrict__ lossOut) {
  const int w = threadIdx.x >> 5, lane = threadIdx.x & 31;
  const int row = blockIdx.x * 8 + w;
  const float* z = zp + (size_t)row * L_DIM;
  float zr[16];
#pragma unroll
  for (int i = 0; i < 16; ++i) zr[i] = z[lane + 32 * i];
  float s[16];
  for (int j = 0; j < 16; ++j) {
    int key = (j == 0) ? kk1[row] : disK[row * DISK + (j - 1)];
    const float* t = Tneg + (size_t)key * L_DIM;
    float p = 0.f;
#pragma unroll
    for (int i = 0; i < 16; ++i) p += zr[i] * t[lane + 32 * i];
    p = wave_sum(p);
    s[j] = p * (1.0f / TEMP);
  }
  float m = s[0];
#pragma unroll
  for (int j = 1; j < 16; ++j) m = fmaxf(m, s[j]);
  float se = 0.f;
#pragma unroll
  for (int j = 0; j < 16; ++j) se += expf(s[j] - m);
  float loss = (m + logf(se)) - s[0];
  if (lane == 0) atomicAdd(lossOut, loss);
}

// ---------------------------------------------------------------------------
extern "C" void kernel_launch(void* const* d_in, const int* in_sizes, int n_in,
                              void* d_out_v, int out_size, void* d_ws, size_t ws_size,
                              hipStream_t stream) {
  (void)in_sizes; (void)n_in; (void)out_size; (void)ws_size;
  const float* logits   = (const float*)d_in[0];
  const int*   rels     = (const int*)d_in[1];
  const float* rel_dict = (const float*)d_in[2];
  const float* eps1 = (const float*)d_in[3];
  const float* eps2 = (const float*)d_in[4];
  const float* W1  = (const float*)d_in[5];
  const float* b1  = (const float*)d_in[6];
  const float* Wmu = (const float*)d_in[7];
  const float* bmu = (const float*)d_in[8];
  const float* Wls = (const float*)d_in[9];
  const float* bls = (const float*)d_in[10];
  const float* W3  = (const float*)d_in[11];
  const float* b3  = (const float*)d_in[12];
  const float* W4  = (const float*)d_in[13];
  const float* b4  = (const float*)d_in[14];
  const float* iW1 = (const float*)d_in[15];
  const float* ib1 = (const float*)d_in[16];
  const float* iW2 = (const float*)d_in[17];
  const float* ib2 = (const float*)d_in[18];
  const float* zW1 = (const float*)d_in[19];
  const float* zb1 = (const float*)d_in[20];
  const float* zW2 = (const float*)d_in[21];
  const float* zb2 = (const float*)d_in[22];
  float* d_out = (float*)d_out_v;

  // workspace layout
  int*   kk1  = (int*)d_ws;
  int*   kk2  = kk1 + B_ROWS;
  int*   disK = kk2 + B_ROWS;
  float* T1a  = (float*)(disK + B_ROWS * DISK);
  float* T1b  = T1a + (size_t)V_DICT * L_DIM;
  float* T3b  = T1b + (size_t)V_DICT * L_DIM;
  float* Tneg = T3b + (size_t)V_DICT * L_DIM;
  float* buf0 = Tneg + (size_t)V_DICT * L_DIM;            // h1 -> z
  float* buf1 = buf0 + (size_t)B_ROWS * L_DIM;            // hi -> mu -> h3
  float* buf2 = buf1 + (size_t)B_ROWS * L_DIM;            // ls -> zh
  float* buf3 = buf2 + (size_t)B_ROWS * L_DIM;            // z_

  init_out_kernel<<<3, 256, 0, stream>>>(d_out);
  topdis_kernel<<<B_ROWS / 8, 256, 0, stream>>>(logits, rels, kk1, kk2, disK);

  const int tabBlocks = V_DICT * L_DIM / 256;
  tab_lin_kernel<<<tabBlocks, 256, 0, stream>>>(rel_dict, W1, 0,     T1a);
  tab_lin_kernel<<<tabBlocks, 256, 0, stream>>>(rel_dict, W1, F_DIM, T1b);
  tab_lin_kernel<<<tabBlocks, 256, 0, stream>>>(rel_dict, W3, L_DIM, T3b);
  imlp1_kernel<<<tabBlocks, 256, 0, stream>>>(rel_dict, iW1, ib1, buf1);  // hi
  imlp2_kernel<<<tabBlocks, 256, 0, stream>>>(buf1, iW2, ib2, Tneg);

  h1_kernel<<<(B_ROWS * L_DIM) / 256, 256, 0, stream>>>(kk1, kk2, T1a, T1b, b1, buf0);

  dim3 g512(B_ROWS / 128, L_DIM / 64);
  gemm512_kernel<0, false><<<g512, 256, 0, stream>>>(buf0, Wmu, bmu, buf1,
      L_DIM, L_DIM, L_DIM, nullptr, nullptr, 0, nullptr);
  gemm512_kernel<0, false><<<g512, 256, 0, stream>>>(buf0, Wls, bls, buf2,
      L_DIM, L_DIM, L_DIM, nullptr, nullptr, 0, nullptr);

  z_kernel<<<(B_ROWS * L_DIM) / 256, 256, 0, stream>>>(buf1, buf2, eps1, eps2,
                                                       buf0, buf3, d_out);  // KL -> d_out[0]

  // h3 = leaky(z @ W3[:512] + T3b[kk2] + b3)
  gemm512_kernel<1, false><<<g512, 256, 0, stream>>>(buf0, W3, b3, buf1,
      L_DIM, L_DIM, L_DIM, kk2, T3b, L_DIM, nullptr);

  // recon loss: sum (tanh(h3@W4+b4) - rel_dict[kk1])^2 -> d_out[0]
  dim3 g300(B_ROWS / 128, (F_DIM + 63) / 64);
  gemm512_kernel<2, true><<<g300, 256, 0, stream>>>(buf1, W4, b4, nullptr,
      F_DIM, 0, L_DIM, kk1, rel_dict, F_DIM, d_out);

  // z-projection MLP; second layer writes straight into z_rel rows 1..B
  gemm512_kernel<3, false><<<g512, 256, 0, stream>>>(buf3, zW1, zb1, buf2,
      L_DIM, L_DIM, L_DIM, nullptr, nullptr, 0, nullptr);
  float* zp = d_out + 2 + L_DIM;
  gemm512_kernel<3, false><<<g512, 256, 0, stream>>>(buf2, zW2, zb2, zp,
      L_DIM, L_DIM, L_DIM, nullptr, nullptr, 0, nullptr);

  csl_kernel<<<B_ROWS / 8, 256, 0, stream>>>(zp, Tneg, kk1, disK, d_out + 1);
}